// SpatialEmbLoss_4990751998185
// MI455X (gfx1250) — compile-verified
//
#include <hip/hip_runtime.h>
#include <hip/hip_bf16.h>

#define HH 512
#define WW 512
#define HW (HH*WW)
#define NIMG 8
#define KINST 16
#define BINS 16384
#define NPAIR (NIMG*KINST)

typedef __attribute__((ext_vector_type(2))) float v2f;
typedef __attribute__((ext_vector_type(8))) float v8f;

// ---- workspace layout (32-bit words) ----
// SUMS:  float[NIMG][16][8]: cnt, sum_x, sum_y, sum_sigma, sum_sigma2, pad...
// STATS: float[NIMG][16][8]: cx, cy, s, s_exp, cnt, present, var, pad
// ACC:   float[NIMG][4]:     inst, var, seed_fg, seed_bg
#define SUMS_OFF   0
#define STATS_OFF  (SUMS_OFF + NIMG*16*8)
#define ACC_OFF    (STATS_OFF + NIMG*16*8)
#define HCNT_OFF   4096                       // u32 [NPAIR][BINS]
#define HGT_OFF    (HCNT_OFF + NPAIR*BINS)    // u32 [NPAIR][BINS]
#define HSUM_OFF   (HGT_OFF + NPAIR*BINS)     // f32 [NPAIR][BINS]
#define WS_WORDS   (HSUM_OFF + NPAIR*BINS)

// ------------------------------------------------------------------
// K1: per-instance moments via one-hot WMMA segmented reduction.
// D[m][n] += sum_k onehot(inst(p_k)==m+1) * feat_n(p_k),  K=4 px/WMMA.
// Features: n=0 -> 1, n=1 -> x, n=2 -> y, n=3 -> sigma, n=4 -> sigma^2.
// Feature select is branchless (per-lane coefficients hoisted out of the
// loop) so the inner loop is shuffles + FMAs + WMMA with EXEC untouched.
// ------------------------------------------------------------------
__global__ __launch_bounds__(256) void k1_moments(const float* __restrict__ pred,
                                                  const int* __restrict__ inst,
                                                  float* __restrict__ ws) {
  const int img  = blockIdx.y;
  const int lane = threadIdx.x & 31;
  const int wave = blockIdx.x * (blockDim.x >> 5) + (threadIdx.x >> 5);
  const int wavesPerImg = gridDim.x * (blockDim.x >> 5);
  const int pixPerWave  = HW / wavesPerImg;      // 262144/256 = 1024
  const int start = wave * pixPerWave;

  const int*   instp = inst + img * HW;
  const float* sigp  = pred + (img * 4 + 2) * HW;
  const float  inv511 = 1.0f / 511.0f;

  const int m     = lane & 15;        // A row id / B feature column
  const int khalf = (lane >> 4) << 1; // lanes 0-15 -> K=0,1 ; lanes 16-31 -> K=2,3

  // per-lane branchless feature coefficients (loop invariant)
  const float c1  = (m == 0) ? 1.0f : 0.0f;
  const float cx  = (m == 1) ? 1.0f : 0.0f;
  const float cy  = (m == 2) ? 1.0f : 0.0f;
  const float cs  = (m == 3) ? 1.0f : 0.0f;
  const float csq = (m == 4) ? 1.0f : 0.0f;
  const int   tgt = m + 1;            // one-hot target instance id

  v8f acc = {};
  for (int base = start; base < start + pixPerWave; base += 32) {
    int   iv = instp[base + lane];    // coalesced 32-pixel loads
    float sv = sigp[base + lane];
#pragma unroll
    for (int g = 0; g < 8; ++g) {     // 8 groups of 4 pixels
      int q0 = g * 4 + khalf;
      int   i0 = __shfl(iv, q0);
      int   i1 = __shfl(iv, q0 + 1);
      float s0 = __shfl(sv, q0);
      float s1 = __shfl(sv, q0 + 1);
      int p0 = base + q0;             // p0 even -> p0, p0+1 share a row
      float x0 = (float)(p0 & 511) * inv511;
      float y0 = (float)(p0 >> 9)  * inv511;
      float x1 = x0 + inv511;

      v2f a, b;
      a.x = (i0 == tgt) ? 1.0f : 0.0f;
      a.y = (i1 == tgt) ? 1.0f : 0.0f;
      b.x = c1 + cx * x0 + cy * y0 + (cs + csq * s0) * s0;
      b.y = c1 + cx * x1 + cy * y0 + (cs + csq * s1) * s1;
      acc = __builtin_amdgcn_wmma_f32_16x16x4_f32(false, a, false, b,
                                                  (short)0, acc, false, false);
    }
  }
  // D layout: lane l holds col (l&15); VGPR v holds row v + 8*(l>=16)
  const int ncol  = lane & 15;
  const int rbase = (lane >> 4) << 3;
  if (ncol < 5) {
#pragma unroll
    for (int v = 0; v < 8; ++v) {
      int r = rbase + v;
      atomicAdd(&ws[SUMS_OFF + ((img * 16 + r) << 3) + ncol], acc[v]);
    }
  }
}

// ------------------------------------------------------------------
// K2: finalize per-instance stats; closed-form variance.
// ------------------------------------------------------------------
__global__ void k2_finalize(float* __restrict__ ws) {
  int t = threadIdx.x;                 // img*16 + k
  if (t >= NIMG * 16) return;
  const float* su = ws + SUMS_OFF + t * 8;
  float cnt = su[0], sx = su[1], sy = su[2], ss = su[3], ss2 = su[4];
  float safe = fmaxf(cnt, 1.0f);
  float s = ss / safe;
  // var = sum((sigma - s)^2 * mask) / (N_SIGMA * safe)
  float var = (ss2 - 2.0f * s * ss + s * s * cnt) / safe;
  float* st = ws + STATS_OFF + t * 8;
  st[0] = sx / safe;
  st[1] = sy / safe;
  st[2] = s;
  st[3] = expf(10.0f * s);
  st[4] = cnt;
  st[5] = (cnt > 0.0f) ? 1.0f : 0.0f;
  st[6] = (cnt > 0.0f) ? var : 0.0f;
  st[7] = 0.0f;
  if (cnt > 0.0f) atomicAdd(&ws[ACC_OFF + (t >> 4) * 4 + 1], var);
}

__device__ __forceinline__ float blockReduce256(float v, float* red) {
  int tid = threadIdx.x;
  red[tid] = v;
  __syncthreads();
  for (int s = 128; s > 0; s >>= 1) {
    if (tid < s) red[tid] += red[tid + s];
    __syncthreads();
  }
  float r = red[0];
  __syncthreads();
  return r;
}

// ------------------------------------------------------------------
// K3: per-pixel pass: dist maps, error histograms, seed/bg accums
// ------------------------------------------------------------------
__global__ __launch_bounds__(256) void k3_pixels(const float* __restrict__ pred,
                                                 const int* __restrict__ inst,
                                                 const int* __restrict__ lab,
                                                 float* __restrict__ ws) {
  __shared__ float st[16 * 8];
  __shared__ float red[256];
  const int img = blockIdx.y;
  const int tid = threadIdx.x;
  if (tid < 128) st[tid] = ws[STATS_OFF + img * 128 + tid];
  __syncthreads();

  const int p = blockIdx.x * 256 + tid;
  const float* pb = pred + (size_t)img * 4 * HW;
  float sx   = tanhf(pb[p])      + (float)(p & 511) * (1.0f / 511.0f);
  float sy   = tanhf(pb[HW + p]) + (float)(p >> 9) * (1.0f / 511.0f);
  float seed = 1.0f / (1.0f + expf(-pb[3 * HW + p]));
  int myinst = inst[img * HW + p];
  int lb     = lab[img * HW + p];

  float bg_part    = (lb == 0) ? seed * seed : 0.0f;
  float seedl_part = 0.0f;

  unsigned* hcnt = (unsigned*)ws + HCNT_OFF;
  unsigned* hgt  = (unsigned*)ws + HGT_OFF;
  float*    hsum = ws + HSUM_OFF;

  for (int k = 0; k < 16; ++k) {
    const float* s8 = &st[k * 8];
    if (s8[5] == 0.0f) continue;     // absent -> zeroed in reference (uniform branch)
    float dx = sx - s8[0], dy = sy - s8[1];
    float d  = expf(-(dx * dx + dy * dy) * s8[3]);
    bool  gt = (myinst == k + 1);
    float e  = gt ? (2.0f - 2.0f * d) : (2.0f * d);  // e in [0,2]; relu is identity
    int bin = (int)(e * (float)(BINS / 2));
    bin = min(max(bin, 0), BINS - 1);
    int idx = (img * 16 + k) * BINS + bin;
    atomicAdd(&hcnt[idx], 1u);
    atomicAdd(&hsum[idx], e);
    if (gt) {
      atomicAdd(&hgt[idx], 1u);
      seedl_part = (seed - d) * (seed - d);          // FG_W = 1
    }
  }

  float sl = blockReduce256(seedl_part, red);
  float bs = blockReduce256(bg_part, red);
  if (tid == 0) {
    atomicAdd(&ws[ACC_OFF + img * 4 + 2], sl);
    atomicAdd(&ws[ACC_OFF + img * 4 + 3], bs);
  }
}

// ------------------------------------------------------------------
// K4: Lovász hinge via descending-bin scan (one block per (img,inst)).
// Within a tie-group: sum e_i*(jac_i - jac_{i-1}) = e*(jac_end - jac_before)
// ------------------------------------------------------------------
__global__ __launch_bounds__(256) void k4_lovasz(float* __restrict__ ws) {
  const int pair = blockIdx.x;     // img*16 + k
  const int img  = pair >> 4;
  const int tid  = threadIdx.x;
  __shared__ float sn[256], sg[256];
  __shared__ float red[256];
  __shared__ float carryN, carryG;

  float present = ws[STATS_OFF + pair * 8 + 5];
  float G       = ws[STATS_OFF + pair * 8 + 4];   // total gt count
  if (present == 0.0f) return;                    // uniform across block

  const unsigned* hcnt = (const unsigned*)ws + HCNT_OFF + pair * BINS;
  const unsigned* hgt  = (const unsigned*)ws + HGT_OFF  + pair * BINS;
  const float*    hsum = ws + HSUM_OFF + (size_t)pair * BINS;

  if (tid == 0) { carryN = 0.0f; carryG = 0.0f; }
  __syncthreads();

  float total = 0.0f;
  for (int c = 0; c < BINS / 256; ++c) {
    int bin = BINS - 1 - (c * 256 + tid);         // descending error order
    float n = (float)hcnt[bin];
    float g = (float)hgt[bin];
    float s = hsum[bin];

    // inclusive Hillis-Steele scan in processing order (tid)
    sn[tid] = n; sg[tid] = g;
    __syncthreads();
    float accN = n, accG = g;
    for (int off = 1; off < 256; off <<= 1) {
      float aN = (tid >= off) ? sn[tid - off] : 0.0f;
      float aG = (tid >= off) ? sg[tid - off] : 0.0f;
      __syncthreads();
      accN += aN; accG += aG;
      sn[tid] = accN; sg[tid] = accG;
      __syncthreads();
    }
    float nBefore = carryN + accN - n;
    float gBefore = carryG + accG - g;
    if (n > 0.0f) {
      float nEnd = nBefore + n, gEnd = gBefore + g;
      // jac(count elems, cum gt) = 1 - (G - cg)/(G + count - cg)
      float jacHi = (nBefore > 0.0f)
                  ? 1.0f - (G - gBefore) / (G + nBefore - gBefore) : 0.0f;
      float jacLo = 1.0f - (G - gEnd) / (G + nEnd - gEnd);
      total += (s / n) * (jacLo - jacHi);          // mean e of bin * grad span
    }
    __syncthreads();
    if (tid == 255) { carryN += accN; carryG += accG; }
    __syncthreads();
  }
  float sum = blockReduce256(total, red);
  if (tid == 0) atomicAdd(&ws[ACC_OFF + img * 4 + 0], sum);
}

// ------------------------------------------------------------------
// K5: combine
// ------------------------------------------------------------------
__global__ void k5_final(const float* __restrict__ ws, float* __restrict__ out) {
  if (threadIdx.x != 0 || blockIdx.x != 0) return;
  float loss = 0.0f;
  for (int img = 0; img < NIMG; ++img) {
    float obj = 0.0f;
    for (int k = 0; k < 16; ++k) obj += ws[STATS_OFF + (img * 16 + k) * 8 + 5];
    float safe = fmaxf(obj, 1.0f);
    const float* a = ws + ACC_OFF + img * 4;
    // W_INST=1, W_VAR=10, W_SEED=1
    loss += (a[0] + 10.0f * a[1]) / safe + (a[2] + a[3]) / (float)HW;
  }
  out[0] = loss / (float)NIMG;
}

extern "C" void kernel_launch(void* const* d_in, const int* in_sizes, int n_in,
                              void* d_out, int out_size, void* d_ws, size_t ws_size,
                              hipStream_t stream) {
  const float* pred = (const float*)d_in[0];   // (8,4,512,512) f32
  const int*   inst = (const int*)d_in[1];     // (8,512,512) i32
  const int*   lab  = (const int*)d_in[2];     // (8,512,512) i32
  float* ws = (float*)d_ws;

  hipMemsetAsync(d_ws, 0, (size_t)WS_WORDS * sizeof(float), stream);

  dim3 g1(32, NIMG);
  k1_moments<<<g1, 256, 0, stream>>>(pred, inst, ws);
  k2_finalize<<<1, 128, 0, stream>>>(ws);
  dim3 g3(HW / 256, NIMG);
  k3_pixels<<<g3, 256, 0, stream>>>(pred, inst, lab, ws);
  k4_lovasz<<<NPAIR, 256, 0, stream>>>(ws);
  k5_final<<<1, 32, 0, stream>>>(ws, (float*)d_out);
}